// NHGNN_17179869472
// MI455X (gfx1250) — compile-verified
//
#include <hip/hip_runtime.h>

#define H 128
#define NHEAD 2
#define NTYPE 13
#define NNODES 50000
#define NEDGES 600000
#define ETILES 4  // edge tiles (of 16) per block in the edge-MLP GEMM

typedef __attribute__((ext_vector_type(2))) float v2f;
typedef __attribute__((ext_vector_type(8))) float v8f;

// ------------------------------------------------------------------
// 0) init: zero in-degree counters and type counters
// ------------------------------------------------------------------
__global__ void k_init(int* __restrict__ deg, int* __restrict__ cnt) {
  int i = blockIdx.x * blockDim.x + threadIdx.x;
  if (i < NNODES) deg[i] = 0;
  if (i < 16) cnt[i] = 0;
}

// ------------------------------------------------------------------
// 1) bucket nodes by type (compaction -> same-type 16-node WMMA tiles)
// ------------------------------------------------------------------
__global__ void k_bucket(const int* __restrict__ ntype, int* __restrict__ cnt,
                         int* __restrict__ bucket) {
  int n = blockIdx.x * blockDim.x + threadIdx.x;
  if (n >= NNODES) return;
  int t = ntype[n];
  int p = atomicAdd(&cnt[t], 1);
  bucket[t * NNODES + p] = n;
}

// ------------------------------------------------------------------
// 2) per-type node linear: x_ = x @ W_node[t]^T + b_node[t]
//    1 block = 16 same-type nodes. 8 waves, wave w -> cols [16w,16w+16).
//    f32 WMMA 16x16x4, K loop of 32. A in LDS (132-float padded stride:
//    bank = (4m+k)%64, conflict-free b64). B fragments: 64 VGPRs/lane.
// ------------------------------------------------------------------
__global__ void __launch_bounds__(256) k_node_wmma(
    const float* __restrict__ x, const float* __restrict__ Wn,
    const float* __restrict__ bn, const int* __restrict__ cnt,
    const int* __restrict__ bucket, float* __restrict__ xw) {
  const int TPT = (NNODES + 15) / 16;  // tiles per type
  int t = blockIdx.x / TPT;
  int tile = blockIdx.x % TPT;
  int c = cnt[t];
  int base = tile * 16;
  if (base >= c) return;  // uniform across block

  __shared__ float sA[16][132];
  __shared__ int sI[16];
  int tid = threadIdx.x;
  if (tid < 16) {
    int slot = base + tid;
    sI[tid] = bucket[t * NNODES + ((slot < c) ? slot : (c - 1))];
  }
  __syncthreads();
  {
    int e = tid >> 4;          // node slot 0..15
    int off = (tid & 15) * 8;  // 8 floats per thread
    int node = sI[e];
    const float* px = x + (long)node * H + off;
#pragma unroll
    for (int j = 0; j < 8; j++) sA[e][off + j] = px[j];
  }
  __syncthreads();

  int lane = tid & 31, wave = tid >> 5;
  int half = lane >> 4, l15 = lane & 15;
  int ng = wave * 16 + l15;                        // output column
  const float* wrow = Wn + ((long)t * H + ng) * H; // W_node[t][ng][*]
  int kb0 = 2 * half;
  v2f b[32];
#pragma unroll
  for (int k = 0; k < 32; k++) b[k] = *(const v2f*)(wrow + k * 4 + kb0);
  v8f acc = {};
#pragma unroll
  for (int k = 0; k < 32; k++) {
    v2f a = *(const v2f*)(&sA[l15][k * 4 + kb0]);
    acc = __builtin_amdgcn_wmma_f32_16x16x4_f32(false, a, false, b[k],
                                                (short)0, acc, false, false);
  }
  float bias = bn[t * H + ng];
#pragma unroll
  for (int r = 0; r < 8; r++) {
    int m = r + 8 * half;
    int slot = base + m;
    if (slot < c) {
      int node = sI[m];
      xw[(long)node * H + ng] = acc[r] + bias;
    }
  }
}

// ------------------------------------------------------------------
// 3) CSR build over destination node (row): degree count
// ------------------------------------------------------------------
__global__ void k_deg(const int* __restrict__ ei, int* __restrict__ deg) {
  int e = blockIdx.x * blockDim.x + threadIdx.x;
  if (e >= NEDGES) return;
  atomicAdd(&deg[ei[e]], 1);
}

// ------------------------------------------------------------------
// 4) single-block prefix sum: rowptr[i+1]=incl, woff[i]=excl scan of deg
// ------------------------------------------------------------------
__global__ void __launch_bounds__(1024) k_scan(const int* __restrict__ deg,
                                               int* __restrict__ rowptr,
                                               int* __restrict__ woff) {
  __shared__ int sbuf[1024];
  __shared__ int scarry;
  int tid = threadIdx.x;
  if (tid == 0) scarry = 0;
  __syncthreads();
  for (int base = 0; base < NNODES; base += 1024) {
    int i = base + tid;
    int v = (i < NNODES) ? deg[i] : 0;
    sbuf[tid] = v;
    __syncthreads();
    for (int ofs = 1; ofs < 1024; ofs <<= 1) {
      int tv = (tid >= ofs) ? sbuf[tid - ofs] : 0;
      __syncthreads();
      sbuf[tid] += tv;
      __syncthreads();
    }
    int inc = sbuf[tid] + scarry;
    if (i < NNODES) {
      rowptr[i + 1] = inc;
      woff[i] = inc - v;  // exclusive prefix
    }
    __syncthreads();
    if (tid == 1023) scarry = inc;
    __syncthreads();
  }
  if (tid == 0) rowptr[0] = 0;
}

// ------------------------------------------------------------------
// 5) scatter edge ids into CSR slots
// ------------------------------------------------------------------
__global__ void k_fill(const int* __restrict__ ei, int* __restrict__ woff,
                       int* __restrict__ elist) {
  int e = blockIdx.x * blockDim.x + threadIdx.x;
  if (e >= NEDGES) return;
  int r = ei[e];
  int p = atomicAdd(&woff[r], 1);
  elist[p] = e;
}

// ------------------------------------------------------------------
// 6) edge scores: s = leaky_relu((x_[row]+x_[col]+ea) @ W_score^T)
//    wave per edge, lane-parallel dot, shfl_xor butterfly reduce.
// ------------------------------------------------------------------
__global__ void __launch_bounds__(256) k_score(
    const float* __restrict__ xw, const int* __restrict__ ei,
    const float* __restrict__ ea, const float* __restrict__ Ws,
    float* __restrict__ score) {
  int wave = threadIdx.x >> 5, lane = threadIdx.x & 31;
  long e = (long)blockIdx.x * 8 + wave;
  if (e >= NEDGES) return;
  int r = ei[e], c = ei[NEDGES + e];
  int off = lane * 4;
  float4 xr = *(const float4*)(xw + (long)r * H + off);
  float4 xc = *(const float4*)(xw + (long)c * H + off);
  float4 a = *(const float4*)(ea + e * H + off);
  float4 w0 = *(const float4*)(Ws + off);
  float4 w1 = *(const float4*)(Ws + H + off);
  float tx = xr.x + xc.x + a.x, ty = xr.y + xc.y + a.y;
  float tz = xr.z + xc.z + a.z, tw = xr.w + xc.w + a.w;
  float p0 = tx * w0.x + ty * w0.y + tz * w0.z + tw * w0.w;
  float p1 = tx * w1.x + ty * w1.y + tz * w1.z + tw * w1.w;
#pragma unroll
  for (int m = 16; m >= 1; m >>= 1) {
    p0 += __shfl_xor(p0, m, 32);
    p1 += __shfl_xor(p1, m, 32);
  }
  if (lane == 0) {
    score[e * 2 + 0] = (p0 >= 0.0f) ? p0 : 0.2f * p0;
    score[e * 2 + 1] = (p1 >= 0.0f) ? p1 : 0.2f * p1;
  }
}

// ------------------------------------------------------------------
// 7) wave-per-node softmax + aggregation (no float atomics):
//    max / sum-exp via shfl reductions over this node's in-edges,
//    then sequential alpha-weighted gather of x_[col]; coalesced
//    float4 store of x_out. Nodes with deg 0 write zeros (matches ref).
// ------------------------------------------------------------------
__global__ void __launch_bounds__(256) k_softagg(
    const int* __restrict__ rowptr, const int* __restrict__ elist,
    const int* __restrict__ ei, const float* __restrict__ score,
    const float* __restrict__ xw, float* __restrict__ xout) {
  int wave = threadIdx.x >> 5, lane = threadIdx.x & 31;
  int n = blockIdx.x * 8 + wave;
  if (n >= NNODES) return;  // wave-uniform
  int beg = rowptr[n], end = rowptr[n + 1];
  int off = lane * 4;
  float4 acc = make_float4(0.f, 0.f, 0.f, 0.f);
  if (end > beg) {  // wave-uniform
    float m0 = -INFINITY, m1 = -INFINITY;
    for (int i = beg + lane; i < end; i += 32) {
      int e = elist[i];
      m0 = fmaxf(m0, score[e * 2 + 0]);
      m1 = fmaxf(m1, score[e * 2 + 1]);
    }
#pragma unroll
    for (int s = 16; s >= 1; s >>= 1) {
      m0 = fmaxf(m0, __shfl_xor(m0, s, 32));
      m1 = fmaxf(m1, __shfl_xor(m1, s, 32));
    }
    float s0 = 0.f, s1 = 0.f;
    for (int i = beg + lane; i < end; i += 32) {
      int e = elist[i];
      s0 += expf(score[e * 2 + 0] - m0);
      s1 += expf(score[e * 2 + 1] - m1);
    }
#pragma unroll
    for (int s = 16; s >= 1; s >>= 1) {
      s0 += __shfl_xor(s0, s, 32);
      s1 += __shfl_xor(s1, s, 32);
    }
    int head = lane >> 4;  // lanes 0-15: dims 0..63 (head0), 16-31: head1
    float mh = head ? m1 : m0;
    float invh = 1.0f / (head ? s1 : s0);
    for (int i = beg; i < end; i++) {
      int e = elist[i];
      int c = ei[NEDGES + e];
      float alpha = expf(score[e * 2 + head] - mh) * invh;
      float4 xc = *(const float4*)(xw + (long)c * H + off);
      acc.x += alpha * xc.x;
      acc.y += alpha * xc.y;
      acc.z += alpha * xc.z;
      acc.w += alpha * xc.w;
    }
  }
  *(float4*)(xout + (long)n * H + off) = acc;
}

// ------------------------------------------------------------------
// 8) edge MLP: edge_out = (x_out[row]+x_out[col]+ea) @ W_edge^T + b_edge
//    Block = 64 edges (4 tiles of 16); B fragments (64 VGPR/lane) loaded
//    once and reused across tiles -> 4x less W_edge refetch traffic.
//    Gather loads issued into registers before the barrier for overlap;
//    prefetch next tile's edge_attr (global_prefetch_b8).
// ------------------------------------------------------------------
__global__ void __launch_bounds__(256) k_edge_wmma(
    const float* __restrict__ xout, const int* __restrict__ ei,
    const float* __restrict__ ea, const float* __restrict__ We,
    const float* __restrict__ be, float* __restrict__ eout) {
  long ebase0 = (long)blockIdx.x * (16 * ETILES);
  __shared__ float sA[16][132];
  int tid = threadIdx.x;
  int lane = tid & 31, wave = tid >> 5;
  int half = lane >> 4, l15 = lane & 15;
  int ng = wave * 16 + l15;
  int kb0 = 2 * half;
  const float* wrow = We + (long)ng * H;  // B[k][n] = W_edge[n][k]
  v2f b[32];
#pragma unroll
  for (int k = 0; k < 32; k++) b[k] = *(const v2f*)(wrow + k * 4 + kb0);
  float bias = be[ng];

  int es = tid >> 4;          // edge slot within tile
  int goff = (tid & 15) * 8;  // 8 floats per thread

  for (int t = 0; t < ETILES; t++) {
    long ebase = ebase0 + (long)t * 16;
    long e = ebase + es;
    int r = ei[e], c = ei[NEDGES + e];
    const float* pr = xout + (long)r * H + goff;
    const float* pc = xout + (long)c * H + goff;
    const float* pa = ea + e * H + goff;
    float v[8];
#pragma unroll
    for (int j = 0; j < 8; j++) v[j] = pr[j] + pc[j] + pa[j];
    if (t + 1 < ETILES) __builtin_prefetch(pa + 16 * H, 0, 1);
    __syncthreads();  // previous tile's WMMA reads of sA are done
#pragma unroll
    for (int j = 0; j < 8; j++) sA[es][goff + j] = v[j];
    __syncthreads();

    v8f acc = {};
#pragma unroll
    for (int k = 0; k < 32; k++) {
      v2f a = *(const v2f*)(&sA[l15][k * 4 + kb0]);
      acc = __builtin_amdgcn_wmma_f32_16x16x4_f32(false, a, false, b[k],
                                                  (short)0, acc, false, false);
    }
#pragma unroll
    for (int r8 = 0; r8 < 8; r8++) {
      int m = r8 + 8 * half;
      eout[(ebase + m) * H + ng] = acc[r8] + bias;
    }
  }
}

// ------------------------------------------------------------------
extern "C" void kernel_launch(void* const* d_in, const int* in_sizes, int n_in,
                              void* d_out, int out_size, void* d_ws,
                              size_t ws_size, hipStream_t stream) {
  const float* x = (const float*)d_in[0];
  const int* ei = (const int*)d_in[1];
  const float* ea = (const float*)d_in[2];
  const int* ntype = (const int*)d_in[3];
  const float* Wn = (const float*)d_in[4];
  const float* bn = (const float*)d_in[5];
  const float* Ws = (const float*)d_in[6];
  const float* We = (const float*)d_in[7];
  const float* be = (const float*)d_in[8];

  float* xout = (float*)d_out;                     // [N, H]
  float* eout = (float*)d_out + (long)NNODES * H;  // [E, H]

  // workspace layout (~36 MB)
  float* xw = (float*)d_ws;                  // N*H floats
  float* score = xw + (long)NNODES * H;      // E*2 floats
  int* cnt = (int*)(score + (long)NEDGES * 2);  // 16
  int* bucket = cnt + 16;                    // NT*N
  int* deg = bucket + NTYPE * NNODES;        // N
  int* woff = deg + NNODES;                  // N
  int* rowptr = woff + NNODES;               // N+1
  int* elist = rowptr + (NNODES + 1);        // E

  k_init<<<(NNODES + 255) / 256, 256, 0, stream>>>(deg, cnt);
  k_bucket<<<(NNODES + 255) / 256, 256, 0, stream>>>(ntype, cnt, bucket);

  const int TPT = (NNODES + 15) / 16;
  k_node_wmma<<<NTYPE * TPT, 256, 0, stream>>>(x, Wn, bn, cnt, bucket, xw);

  k_deg<<<(NEDGES + 255) / 256, 256, 0, stream>>>(ei, deg);
  k_scan<<<1, 1024, 0, stream>>>(deg, rowptr, woff);
  k_fill<<<(NEDGES + 255) / 256, 256, 0, stream>>>(ei, woff, elist);

  k_score<<<(NEDGES + 7) / 8, 256, 0, stream>>>(xw, ei, ea, Ws, score);
  k_softagg<<<(NNODES + 7) / 8, 256, 0, stream>>>(rowptr, elist, ei, score, xw,
                                                  xout);
  k_edge_wmma<<<NEDGES / (16 * ETILES), 256, 0, stream>>>(xout, ei, ea, We, be,
                                                          eout);
}